// GNNEncoder_67585605370471
// MI455X (gfx1250) — compile-verified
//
#include <hip/hip_runtime.h>

typedef __attribute__((ext_vector_type(2))) float v2f;
typedef __attribute__((ext_vector_type(8))) float v8f;

#define D       128
#define MBLK    64     // node rows per block
#define NTHR    512    // 16 waves
#define FSTR    132    // padded f32 LDS row stride (528B -> 4-bank rotation, 8B aligned)

__global__ void gin_zero_kernel(float* __restrict__ p, int n) {
    int i = blockIdx.x * blockDim.x + threadIdx.x;
    if (i < n) p[i] = 0.0f;
}

// One wave32 per edge: lanes cover the 128-float row (float4 each), 4 f32 atomics per lane.
__global__ void gin_scatter_kernel(const float* __restrict__ h,
                                   const int* __restrict__ src,
                                   const int* __restrict__ dst,
                                   float* __restrict__ agg, int nEdges) {
    int gid  = blockIdx.x * blockDim.x + threadIdx.x;
    int e    = gid >> 5;
    if (e >= nEdges) return;
    int lane = gid & 31;
    int s = src[e], d = dst[e];
    const float4 v = *(const float4*)(h + (size_t)s * D + lane * 4);
    float* a = agg + (size_t)d * D + lane * 4;
    atomicAdd(a + 0, v.x);
    atomicAdd(a + 1, v.y);
    atomicAdd(a + 2, v.z);
    atomicAdd(a + 3, v.w);
}

// Fused, full fp32 via V_WMMA_F32_16X16X4_F32:
//   z = h+agg; h1 = relu(z@W1+b1); z2 = h1@W2+b2; out = relu(LN(z2))
__global__ __launch_bounds__(NTHR)
void gin_mlp_ln_kernel(const float* __restrict__ h,   const float* __restrict__ agg,
                       const float* __restrict__ W1,  const float* __restrict__ b1,
                       const float* __restrict__ W2,  const float* __restrict__ b2,
                       const float* __restrict__ gamma, const float* __restrict__ beta,
                       float* __restrict__ out, int nNodes) {
    extern __shared__ char smem[];
    float* sW  = (float*)smem;          // [128][FSTR]  W^T f32 (W1 then W2)
    float* sZ  = sW + D * FSTR;         // [64][FSTR]   z f32
    float* sH1 = sZ + MBLK * FSTR;      // [64][FSTR]   relu(z@W1+b1) f32
    float* sO  = sZ;                    // [64][FSTR]   aliases sZ (dead after GEMM1)

    const int tid  = threadIdx.x;
    const int row0 = blockIdx.x * MBLK;

    // Stage W1^T (each output column n contiguous in k) and z = h + agg.
    for (int idx = tid; idx < D * D; idx += NTHR) {
        int k = idx >> 7, n = idx & (D - 1);
        sW[n * FSTR + k] = W1[idx];
    }
    for (int idx = tid; idx < MBLK * D; idx += NTHR) {
        int r = idx >> 7, c = idx & (D - 1);
        int row = row0 + r;
        float v = 0.0f;
        if (row < nNodes) { size_t o = (size_t)row * D + c; v = h[o] + agg[o]; }
        sZ[r * FSTR + c] = v;
    }
    __syncthreads();

    const int wave = tid >> 5;
    const int lane = tid & 31;
    const int mt   = wave >> 2;            // 4 M-tiles of 16 rows
    const int nt0  = (wave & 3) * 2;       // each wave: 2 N-tiles of 16 cols
    const int l16  = lane & 15;
    const int hi   = lane >> 4;            // lane half: K pair 0/1 vs 2/3

    const int arow = mt * 16 + l16;
    const int n0   = nt0 * 16 + l16;
    const int n1   = n0 + 16;

    // ---------------- GEMM1: sZ @ W1 (f32 WMMA, K=4 per step) ----------------
    v8f acc0 = {}; v8f acc1 = {};
    {
        const float* zr = sZ + arow * FSTR;
        const float* w0 = sW + n0 * FSTR;
        const float* w1 = sW + n1 * FSTR;
#pragma unroll
        for (int ks = 0; ks < 32; ++ks) {
            int kk = ks * 4 + hi * 2;
            v2f a  = *(const v2f*)(zr + kk);
            v2f bA = *(const v2f*)(w0 + kk);
            v2f bB = *(const v2f*)(w1 + kk);
            acc0 = __builtin_amdgcn_wmma_f32_16x16x4_f32(false, a, false, bA,
                                                         (short)0, acc0, false, false);
            acc1 = __builtin_amdgcn_wmma_f32_16x16x4_f32(false, a, false, bB,
                                                         (short)0, acc1, false, false);
        }
    }
    // bias + ReLU -> sH1 (f32)
    {
        float bb0 = b1[n0], bb1 = b1[n1];
#pragma unroll
        for (int m = 0; m < 8; ++m) {
            int rm = mt * 16 + hi * 8 + m;
            float v0 = acc0[m] + bb0; v0 = v0 > 0.0f ? v0 : 0.0f;
            float v1 = acc1[m] + bb1; v1 = v1 > 0.0f ? v1 : 0.0f;
            sH1[rm * FSTR + n0] = v0;
            sH1[rm * FSTR + n1] = v1;
        }
    }
    __syncthreads();

    // Stage W2^T over sW (W1 data dead).
    for (int idx = tid; idx < D * D; idx += NTHR) {
        int k = idx >> 7, n = idx & (D - 1);
        sW[n * FSTR + k] = W2[idx];
    }
    __syncthreads();

    // ---------------- GEMM2: sH1 @ W2 ----------------
    v8f d0 = {}; v8f d1 = {};
    {
        const float* hr = sH1 + arow * FSTR;
        const float* w0 = sW + n0 * FSTR;
        const float* w1 = sW + n1 * FSTR;
#pragma unroll
        for (int ks = 0; ks < 32; ++ks) {
            int kk = ks * 4 + hi * 2;
            v2f a  = *(const v2f*)(hr + kk);
            v2f bA = *(const v2f*)(w0 + kk);
            v2f bB = *(const v2f*)(w1 + kk);
            d0 = __builtin_amdgcn_wmma_f32_16x16x4_f32(false, a, false, bA,
                                                       (short)0, d0, false, false);
            d1 = __builtin_amdgcn_wmma_f32_16x16x4_f32(false, a, false, bB,
                                                       (short)0, d1, false, false);
        }
    }
    // bias -> f32 staging (aliases sZ; only sH1 was read by GEMM2, so no hazard)
    {
        float bb0 = b2[n0], bb1 = b2[n1];
#pragma unroll
        for (int m = 0; m < 8; ++m) {
            int rm = mt * 16 + hi * 8 + m;
            sO[rm * FSTR + n0] = d0[m] + bb0;
            sO[rm * FSTR + n1] = d1[m] + bb1;
        }
    }
    __syncthreads();

    // ---------------- LayerNorm + ReLU ----------------
    if (tid < MBLK) {
        int row = row0 + tid;
        if (row < nNodes) {
            const float* r = sO + tid * FSTR;
            float mu = 0.0f;
#pragma unroll 8
            for (int j = 0; j < D; ++j) mu += r[j];
            mu *= (1.0f / D);
            float var = 0.0f;
#pragma unroll 8
            for (int j = 0; j < D; ++j) { float dd = r[j] - mu; var += dd * dd; }
            var *= (1.0f / D);
            float is = rsqrtf(var + 1e-5f);
            float* o = out + (size_t)row * D;
#pragma unroll 4
            for (int j = 0; j < D; ++j) {
                float v = (r[j] - mu) * is * gamma[j] + beta[j];
                o[j] = v > 0.0f ? v : 0.0f;
            }
        }
    }
}

extern "C" void kernel_launch(void* const* d_in, const int* in_sizes, int n_in,
                              void* d_out, int out_size, void* d_ws, size_t ws_size,
                              hipStream_t stream) {
    const float* x     = (const float*)d_in[0];
    const int*   ei    = (const int*)d_in[1];
    const float* W1    = (const float*)d_in[2];
    const float* b1    = (const float*)d_in[3];
    const float* W2    = (const float*)d_in[4];
    const float* b2    = (const float*)d_in[5];
    const float* gamma = (const float*)d_in[6];
    const float* beta  = (const float*)d_in[7];

    const int N = in_sizes[0] / D;
    const int E = in_sizes[1] / 2;
    const int L = 3;

    const int* src = ei;
    const int* dst = ei + E;

    float* out = (float*)d_out;
    float* agg = (float*)d_ws;          // N*D floats of scratch

    const size_t smemBytes = (size_t)(D * FSTR + 2 * MBLK * FSTR) * sizeof(float);

    const int zeroBlocks    = (N * D + 1023) / 1024;
    const int scatterBlocks = (E * 32 + 255) / 256;
    const int mlpBlocks     = (N + MBLK - 1) / MBLK;

    const float* hcur = x;
    for (int l = 0; l < L; ++l) {
        gin_zero_kernel<<<zeroBlocks, 1024, 0, stream>>>(agg, N * D);
        gin_scatter_kernel<<<scatterBlocks, 256, 0, stream>>>(hcur, src, dst, agg, E);
        gin_mlp_ln_kernel<<<mlpBlocks, NTHR, smemBytes, stream>>>(
            hcur, agg,
            W1 + (size_t)l * D * D, b1 + (size_t)l * D,
            W2 + (size_t)l * D * D, b2 + (size_t)l * D,
            gamma + (size_t)l * D, beta + (size_t)l * D,
            out, N);
        hcur = out;   // in-place safe: each block reads only the rows it writes
    }
}